// LiquidNet2_27230092657496
// MI455X (gfx1250) — compile-verified
//
#include <hip/hip_runtime.h>
#include <cstdint>

#define AS1 __attribute__((address_space(1)))
#define AS3 __attribute__((address_space(3)))

#ifndef __has_builtin
#define __has_builtin(x) 0
#endif

// vector types matching the async-LDS builtin prototypes
typedef int vi4 __attribute__((vector_size(4 * sizeof(int))));
typedef int vi2 __attribute__((vector_size(2 * sizeof(int))));

// ---------------- problem constants ----------------
constexpr int Bv = 1024;   // batch
constexpr int Iv = 128;    // input dim
constexpr int Sv = 512;    // state dim
constexpr int BT = 32;     // batch tile per block (2 rows per thread)
constexpr int JT = 16;     // state-column tile per block
constexpr int KC = 32;     // presynaptic chunk staged in LDS
constexpr int VPAD = 34;   // padded vbuf row (bank-conflict free, 8B aligned)

// ---------------- async global->LDS helpers (CDNA5 ASYNCcnt path) ----------------
__device__ __forceinline__ void async_ld16(const void* g, void* l) {
#if __has_builtin(__builtin_amdgcn_global_load_async_to_lds_b128)
  __builtin_amdgcn_global_load_async_to_lds_b128((AS1 vi4*)g, (AS3 vi4*)l, 0, 0);
#else
  uint32_t laddr = (uint32_t)(uintptr_t)(AS3 void*)l;
  asm volatile("global_load_async_to_lds_b128 %0, %1, off"
               :: "v"(laddr), "v"((uint64_t)(uintptr_t)g) : "memory");
#endif
}

__device__ __forceinline__ void async_ld8(const void* g, void* l) {
#if __has_builtin(__builtin_amdgcn_global_load_async_to_lds_b64)
  __builtin_amdgcn_global_load_async_to_lds_b64((AS1 vi2*)g, (AS3 vi2*)l, 0, 0);
#else
  uint32_t laddr = (uint32_t)(uintptr_t)(AS3 void*)l;
  asm volatile("global_load_async_to_lds_b64 %0, %1, off"
               :: "v"(laddr), "v"((uint64_t)(uintptr_t)g) : "memory");
#endif
}

template <int N>
__device__ __forceinline__ void wait_asynccnt() {
#if __has_builtin(__builtin_amdgcn_s_wait_asynccnt)
  __builtin_amdgcn_s_wait_asynccnt((unsigned short)N);
#else
  asm volatile("s_wait_asynccnt %0" :: "i"(N) : "memory");
#endif
}

// gfx1250 hardware tanh (V_TANH_F32); exp2/rcp fallback keeps identical packing
__device__ __forceinline__ float fast_tanh(float x) {
#if __has_builtin(__builtin_amdgcn_tanhf)
  return __builtin_amdgcn_tanhf(x);
#else
  float e = __builtin_amdgcn_exp2f(x * 2.8853900817779268f);  // 2*log2e
  return 1.0f - 2.0f * __builtin_amdgcn_rcpf(e + 1.0f);
#endif
}

// ---------------- shared synapse accumulation body ----------------
// sigmoid(sigma*(v-mu)) = 0.5 + 0.5*tanh(0.5*sigma*(v-mu)); the 0.5 constants
// are hoisted into batch-independent column sums, so the hot loop per element
// is 3 FMA + 1 v_tanh_f32 on LDS-staged (async global->LDS) tiles.
__device__ __forceinline__ void synapse_accum(
    const float4* __restrict__ P, const float* __restrict__ act, int K,
    int j0, int b0, float4 (*pbuf)[KC * JT], float (*vbuf)[BT * VPAD],
    float& num0, float& den0, float& num1, float& den1) {
  const int t   = (int)threadIdx.x;
  const int jl  = t & 15;
  const int bl  = t >> 4;
  const int row = t >> 4;
  const int col = t & 15;
  const int NC  = K / KC;

  auto issue = [&](int c, int sel) {
    const int i0 = c * KC;
    // 8KB param tile: 2 x (256 lanes x 16B)
    async_ld16(&P[(size_t)(i0 + row) * Sv + (j0 + col)],
               &pbuf[sel][row * JT + col]);
    async_ld16(&P[(size_t)(i0 + 16 + row) * Sv + (j0 + col)],
               &pbuf[sel][(16 + row) * JT + col]);
    // 4KB activation tile: 2 x (256 lanes x 8B), padded LDS rows
    async_ld8(&act[(size_t)(b0 + row) * K + i0 + col * 2],
              &vbuf[sel][row * VPAD + col * 2]);
    async_ld8(&act[(size_t)(b0 + 16 + row) * K + i0 + col * 2],
              &vbuf[sel][(16 + row) * VPAD + col * 2]);
  };

  issue(0, 0);
  for (int c = 0; c < NC; ++c) {
    const int sel = c & 1;
    issue((c + 1 == NC) ? 0 : (c + 1), sel ^ 1);  // double-buffer prefetch
    wait_asynccnt<4>();                           // chunk c landed (in-order)
    __syncthreads();                              // cross-wave visibility
#pragma unroll
    for (int ii = 0; ii < KC; ++ii) {
      float  vv0 = vbuf[sel][bl * VPAD + ii];          // broadcast across j-lanes
      float  vv1 = vbuf[sel][(bl + 16) * VPAD + ii];
      float4 p   = pbuf[sel][ii * JT + jl];            // 1 float4 feeds 2 batches
      float th0 = fast_tanh(fmaf(p.x, vv0, p.y));      // tanh(0.5*sigma*(v-mu))
      float th1 = fast_tanh(fmaf(p.x, vv1, p.y));
      num0 = fmaf(p.w, th0, num0);
      den0 = fmaf(p.z, th0, den0);
      num1 = fmaf(p.w, th1, num1);
      den1 = fmaf(p.z, th1, den1);
    }
    __syncthreads();                              // before buffer reuse
  }
}

// ---------------- recurrent unfold kernel (defined FIRST for disasm view) ----
__global__ __launch_bounds__(256) void synapse_rec_kernel(
    const float4* __restrict__ P,    // [Sv, Sv] packed {a,c,wh,wnh}
    const float*  __restrict__ act,  // [Bv, Sv] current state v
    const float* __restrict__ csn, const float* __restrict__ csd,
    const float* __restrict__ wns, const float* __restrict__ wds,
    const float* __restrict__ cm,  const float* __restrict__ gl,
    const float* __restrict__ vl,
    float* __restrict__ out0) {
  __shared__ float4 pbuf[2][KC * JT];
  __shared__ float  vbuf[2][BT * VPAD];
  const int t  = (int)threadIdx.x;
  const int jl = t & 15, bl = t >> 4;
  const int j0 = (int)blockIdx.x * JT;
  const int b0 = (int)blockIdx.y * BT;

  float num0 = 0.0f, den0 = 0.0f, num1 = 0.0f, den1 = 0.0f;
  synapse_accum(P, act, Sv, j0, b0, pbuf, vbuf, num0, den0, num1, den1);

  const int j = j0 + jl;
  const int ba = b0 + bl, bb = b0 + 16 + bl;
  const float cn = csn[j], cd = csd[j];
  float cmj = cm[j], glj = gl[j];
  float leak = glj * vl[j];
  float voa = act[(size_t)ba * Sv + j];
  float vob = act[(size_t)bb * Sv + j];
  float numa = fmaf(cmj, voa, leak + wns[(size_t)ba * Sv + j] + num0 + cn);
  float dena = cmj + glj + wds[(size_t)ba * Sv + j] + den0 + cd;
  float numb = fmaf(cmj, vob, leak + wns[(size_t)bb * Sv + j] + num1 + cn);
  float denb = cmj + glj + wds[(size_t)bb * Sv + j] + den1 + cd;
  out0[(size_t)ba * Sv + j] = numa / dena;
  out0[(size_t)bb * Sv + j] = numb / denb;
}

// ---------------- sensory kernel ----------------
__global__ __launch_bounds__(256) void synapse_sens_kernel(
    const float4* __restrict__ P,    // [Iv, Sv] packed
    const float*  __restrict__ act,  // [Bv, Iv] affine-mapped inputs x
    const float* __restrict__ csn, const float* __restrict__ csd,
    float* __restrict__ out0, float* __restrict__ out1) {
  __shared__ float4 pbuf[2][KC * JT];
  __shared__ float  vbuf[2][BT * VPAD];
  const int t  = (int)threadIdx.x;
  const int jl = t & 15, bl = t >> 4;
  const int j0 = (int)blockIdx.x * JT;
  const int b0 = (int)blockIdx.y * BT;

  float num0 = 0.0f, den0 = 0.0f, num1 = 0.0f, den1 = 0.0f;
  synapse_accum(P, act, Iv, j0, b0, pbuf, vbuf, num0, den0, num1, den1);

  const int j = j0 + jl;
  const int ba = b0 + bl, bb = b0 + 16 + bl;
  const float cn = csn[j], cd = csd[j];
  out0[(size_t)ba * Sv + j] = num0 + cn;
  out1[(size_t)ba * Sv + j] = den0 + cd;
  out0[(size_t)bb * Sv + j] = num1 + cn;
  out1[(size_t)bb * Sv + j] = den1 + cd;
}

// ---------------- preprocessing kernels ----------------
__global__ __launch_bounds__(256) void pack_params_kernel(
    const float* __restrict__ mu, const float* __restrict__ sigma,
    const float* __restrict__ W,  const float* __restrict__ erev,
    float4* __restrict__ P, int n) {
  int t = blockIdx.x * blockDim.x + threadIdx.x;
  if (t < n) {
    float sg = sigma[t], m = mu[t], w = W[t], er = erev[t];
    float a = 0.5f * sg;
    P[t] = make_float4(a, -a * m, 0.5f * w, 0.5f * w * er);
  }
}

// Column sums of the packed halves: csn[j] = sum_i wnh, csd[j] = sum_i wh
__global__ __launch_bounds__(256) void colsum_kernel(
    const float4* __restrict__ P, int K,
    float* __restrict__ csn, float* __restrict__ csd) {
  int j = blockIdx.x * blockDim.x + threadIdx.x;
  if (j < Sv) {
    float sn = 0.0f, sd = 0.0f;
    for (int i = 0; i < K; ++i) {
      float4 p = P[(size_t)i * Sv + j];
      sn += p.w;
      sd += p.z;
    }
    csn[j] = sn;
    csd[j] = sd;
  }
}

__global__ __launch_bounds__(256) void affine_x_kernel(
    const float* __restrict__ in, const float* __restrict__ w,
    const float* __restrict__ b, float* __restrict__ x, int n, int icols) {
  int t = blockIdx.x * blockDim.x + threadIdx.x;
  if (t < n) {
    int i = t % icols;
    x[t] = fmaf(in[t], w[i], b[i]);
  }
}

// ---------------- host launcher ----------------
extern "C" void kernel_launch(void* const* d_in, const int* in_sizes, int n_in,
                              void* d_out, int out_size, void* d_ws, size_t ws_size,
                              hipStream_t stream) {
  (void)in_sizes; (void)n_in; (void)out_size; (void)ws_size;
  const float* inputs  = (const float*)d_in[0];
  const float* hx      = (const float*)d_in[1];
  const float* input_w = (const float*)d_in[2];
  const float* input_b = (const float*)d_in[3];
  const float* s_mu    = (const float*)d_in[4];
  const float* s_sig   = (const float*)d_in[5];
  const float* s_W     = (const float*)d_in[6];
  const float* s_erev  = (const float*)d_in[7];
  const float* mu      = (const float*)d_in[8];
  const float* sigma   = (const float*)d_in[9];
  const float* W       = (const float*)d_in[10];
  const float* erev    = (const float*)d_in[11];
  const float* vleak   = (const float*)d_in[12];
  const float* gleak   = (const float*)d_in[13];
  const float* cm_t    = (const float*)d_in[14];

  float*  ws    = (float*)d_ws;
  float4* Psens = (float4*)ws;                                   // I*S float4  (1MB)
  float4* Prec  = (float4*)(ws + (size_t)Iv * Sv * 4);           // S*S float4  (4MB)
  float*  x     = ws + (size_t)Iv * Sv * 4 + (size_t)Sv * Sv * 4;
  float*  wns   = x    + (size_t)Bv * Iv;
  float*  wds   = wns  + (size_t)Bv * Sv;
  float*  vA    = wds  + (size_t)Bv * Sv;
  float*  vB    = vA   + (size_t)Bv * Sv;
  float*  csSn  = vB   + (size_t)Bv * Sv;                        // 4 x 512 col sums
  float*  csSd  = csSn + Sv;
  float*  csRn  = csSd + Sv;
  float*  csRd  = csRn + Sv;                                     // total ~13.5MB

  pack_params_kernel<<<(Iv * Sv + 255) / 256, 256, 0, stream>>>(
      s_mu, s_sig, s_W, s_erev, Psens, Iv * Sv);
  pack_params_kernel<<<(Sv * Sv + 255) / 256, 256, 0, stream>>>(
      mu, sigma, W, erev, Prec, Sv * Sv);
  colsum_kernel<<<(Sv + 255) / 256, 256, 0, stream>>>(Psens, Iv, csSn, csSd);
  colsum_kernel<<<(Sv + 255) / 256, 256, 0, stream>>>(Prec, Sv, csRn, csRd);
  affine_x_kernel<<<(Bv * Iv + 255) / 256, 256, 0, stream>>>(
      inputs, input_w, input_b, x, Bv * Iv, Iv);

  dim3 grid(Sv / JT, Bv / BT);  // 32 x 32 blocks, 256 threads each

  // Sensory pass (once): num/den contributions summed over I
  synapse_sens_kernel<<<grid, 256, 0, stream>>>(Psens, x, csSn, csSd, wns, wds);

  // 6 ODE unfolds, ping-ponged; last writes d_out
  const float* vin = hx;
  float* bufs[2] = {vA, vB};
  for (int u = 0; u < 6; ++u) {
    float* vout = (u == 5) ? (float*)d_out : bufs[u & 1];
    synapse_rec_kernel<<<grid, 256, 0, stream>>>(
        Prec, vin, csRn, csRd, wns, wds, cm_t, gleak, vleak, vout);
    vin = vout;
  }
}